// SMILESConstraintLayer_3427383902409
// MI455X (gfx1250) — compile-verified
//
#include <hip/hip_runtime.h>
#include <stdint.h>

#define NEGV -1000000000.0f

// token ids from the reference
#define T_OPEN_P  0
#define T_CLOSE_P 1
#define T_OPEN_S  2
#define T_CLOSE_S 3
#define T_GT      4
#define T_PCT     5
#define T_EQ      6
#define T_HASH    7
#define T_C       8
#define T_O       9
#define T_N       10
#define T_D0      11

#define WAVES_PER_BLOCK 8   // 8 rows per 256-thread block

__global__ __launch_bounds__(256) void smiles_mask_kernel(
    const float* __restrict__ logits,
    const void*  __restrict__ tokens,   // int64 tokens, values 0..63
    float*       __restrict__ out,
    int nrows)
{
    // 8 waves * 4KB staging per row
    __shared__ uint4 lds[WAVES_PER_BLOCK * 256];

    const int wave = threadIdx.x >> 5;
    const int lane = threadIdx.x & 31;
    const int row  = blockIdx.x * WAVES_PER_BLOCK + wave;
    if (row >= nrows) return;

    // ---------------- stage row (512 x int64 = 4KB) global -> LDS, async ----------------
    // Token stream is 128 MB read exactly once: mark it non-temporal so it does
    // not evict useful lines from the 192 MB L2 (TH=NT, load policy 1).
    const unsigned long long gbase = (unsigned long long)tokens;          // uniform SGPR pair
    const uint32_t lds_wave_base = (uint32_t)(uintptr_t)(&lds[wave * 256]);
    const uint32_t row_byte = (uint32_t)row * 4096u;                      // < 128MB, fits 32b
#pragma unroll
    for (int k = 0; k < 8; ++k) {
        uint32_t local_off = (uint32_t)(k * 512 + lane * 16);
        uint32_t voff    = row_byte + local_off;       // 32-bit offset, GVS mode
        uint32_t ldsaddr = lds_wave_base + local_off;  // LDS byte address
        asm volatile("global_load_async_to_lds_b128 %0, %1, %2 th:TH_LOAD_NT"
                     :: "v"(ldsaddr), "v"(voff), "s"(gbase) : "memory");
    }
    asm volatile("s_wait_asynccnt 0x0" ::: "memory");

    // ---------------- each lane owns 16 contiguous tokens ----------------
    const uint4* chunk = &lds[wave * 256 + lane * 8];
    uint32_t tk[16];
#pragma unroll
    for (int k = 0; k < 8; ++k) {
        uint4 q = chunk[k];          // two int64 tokens; low dwords hold the value
        tk[2 * k]     = q.x;
        tk[2 * k + 1] = q.z;
    }

    // broadcast last token of the row (lane 31, element 15)
    const uint32_t last = (uint32_t)__shfl((int)tk[15], 31);

    // ---- lane-local: (sum, min-inclusive-prefix) of paren deltas + ring pairs ----
    int s = 0;
    int m = 0x7fffffff;
    bool hit = false;
#pragma unroll
    for (int i = 0; i < 16; ++i) {
        uint32_t t = tk[i];
        int d = ((t == T_OPEN_P) | (t == T_OPEN_S)) ? 1
              : (((t == T_CLOSE_P) | (t == T_CLOSE_S)) ? -1 : 0);
        s += d;
        m = min(m, s);
        if (i < 15) hit |= (tk[i] == last) & (tk[i + 1] == T_PCT);
    }
    // cross-lane pair (tk[15], next lane's tk[0]); pair j=511 does not exist
    uint32_t nxt0 = (uint32_t)__shfl_down((int)tk[0], 1);
    if (lane < 31) hit |= (tk[15] == last) & (nxt0 == T_PCT);

    // ---- wave32 inclusive scan of per-lane sums ----
    int ps = s;
#pragma unroll
    for (int off = 1; off < 32; off <<= 1) {
        int n = __shfl_up(ps, off);
        if (lane >= off) ps += n;
    }
    const int total = __shfl(ps, 31);       // full-row sum
    int cand = (ps - s) + m;                // global-based min prefix within my chunk
#pragma unroll
    for (int off = 16; off >= 1; off >>= 1)
        cand = min(cand, __shfl_xor(cand, off));
    const int  gmin       = cand;           // min over all 512 inclusive prefix sums
    const bool unbalanced = (total - min(gmin, 0)) > 0;

    const bool anyhit    = (__ballot(hit) != 0ull);
    const bool ring_cond = (last >= T_D0) && (last <= T_D0 + 9) && anyhit;

    // ---- bond count over tail tokens 509..511 = lane 31's tk[13..15] ----
    int bc = 0;
    bc += (tk[13] == T_EQ) | (tk[13] == T_HASH);
    bc += (tk[14] == T_EQ) | (tk[14] == T_HASH);
    bc += (tk[15] == T_EQ) | (tk[15] == T_HASH);
    const int bcount = __shfl(bc, 31);
    const int maxb = (last == T_C) ? 4 : (last == T_O) ? 2 : (last == T_N) ? 3 : 99;
    const bool val_cond = bcount >= maxb;

    // ---- masked add onto logits: two contiguous columns per lane (b64) ----
    const float2 lg = ((const float2*)logits)[(size_t)row * 32 + lane];
    const int v0 = lane * 2, v1 = v0 + 1;
    const bool m0 = (v0 == T_GT && unbalanced) ||
                    ((v0 == T_EQ || v0 == T_HASH) && val_cond) ||
                    (ring_cond && v0 == (int)last);
    const bool m1 = (v1 == T_GT && unbalanced) ||
                    ((v1 == T_EQ || v1 == T_HASH) && val_cond) ||
                    (ring_cond && v1 == (int)last);
    float2 r;
    r.x = lg.x + (m0 ? NEGV : 0.0f);
    r.y = lg.y + (m1 ? NEGV : 0.0f);
    ((float2*)out)[(size_t)row * 32 + lane] = r;
}

extern "C" void kernel_launch(void* const* d_in, const int* in_sizes, int n_in,
                              void* d_out, int out_size, void* d_ws, size_t ws_size,
                              hipStream_t stream)
{
    const float* logits = (const float*)d_in[0];
    const void*  tokens = d_in[1];           // int64 previous_tokens
    float* out = (float*)d_out;

    const int B = in_sizes[0] / 64;          // rows (V = 64)
    const int blocks = (B + WAVES_PER_BLOCK - 1) / WAVES_PER_BLOCK;
    smiles_mask_kernel<<<blocks, 256, 0, stream>>>(logits, tokens, out, B);
}